// TSPM_topKSelection_755914244102
// MI455X (gfx1250) — compile-verified
//
#include <hip/hip_runtime.h>

typedef __attribute__((ext_vector_type(2))) float v2f;
typedef __attribute__((ext_vector_type(8))) float v8f;

#define B_   16
#define T_   8192
#define C_   512
#define H_   4
#define D_   128
#define TOPK 10

// ---------------- Kernel 1: q = qst @ Wq^T + bq   (16 x 512) ----------------
__global__ void k_qproj(const float* __restrict__ qst, const float* __restrict__ Wq,
                        const float* __restrict__ bq, float* __restrict__ q) {
    int i = blockIdx.x * blockDim.x + threadIdx.x;   // 0..8191
    int b = i / C_, c = i % C_;
    const float* x = qst + b * C_;
    const float* w = Wq + (size_t)c * C_;            // row c of Wq (Wq.T column)
    float acc = bq[c];
    for (int j = 0; j < C_; ++j) acc = fmaf(x[j], w[j], acc);
    q[i] = acc;
}

// ------------- Kernel 2: w[b,c,h] = sum_d q[b,hD+d] * Wk[hD+d, c] -----------
__global__ void k_fold(const float* __restrict__ q, const float* __restrict__ Wk,
                       float* __restrict__ wmat) {
    int i = blockIdx.x * blockDim.x + threadIdx.x;   // 0..32767
    int h = i % H_;
    int c = (i / H_) % C_;
    int b = i / (H_ * C_);
    float acc = 0.f;
    const float* qh = q + b * C_ + h * D_;
    for (int d = 0; d < D_; ++d)
        acc = fmaf(qh[d], Wk[(size_t)(h * D_ + d) * C_ + c], acc);
    wmat[(size_t)(b * C_ + c) * H_ + h] = acc;       // layout [b][c][h]
}

// ---------------- Kernel 3: qbk[b,h] = q[b,h,:] . bk[hD:hD+D] ---------------
__global__ void k_qbk(const float* __restrict__ q, const float* __restrict__ bk,
                      float* __restrict__ qbk) {
    int i = threadIdx.x;
    if (i >= B_ * H_) return;
    int h = i % H_, b = i / H_;
    float acc = 0.f;
    for (int d = 0; d < D_; ++d) acc = fmaf(q[b * C_ + h * D_ + d], bk[h * D_ + d], acc);
    qbk[i] = acc;
}

// --------- Kernel 4 (main, WMMA f32): scores[b,h,t] via visual x w ----------
// grid = (T_/128, B_), block = 256 (8 waves, 16 t-rows each)
__global__ void k_scores(const float* __restrict__ visual, const float* __restrict__ wmat,
                         const float* __restrict__ qbk, float* __restrict__ scores) {
    __shared__ float wlds[C_][16];    // B-matrix staging: N padded 4 -> 16, zero-filled
    const int b    = blockIdx.y;
    const int tid  = threadIdx.x;

    for (int i = tid; i < C_ * 16; i += blockDim.x) wlds[i >> 4][i & 15] = 0.f;
    __syncthreads();
    for (int i = tid; i < C_ * H_; i += blockDim.x) {
        int c = i / H_, h = i % H_;
        wlds[c][h] = wmat[(size_t)(b * C_ + c) * H_ + h];
    }
    __syncthreads();

    const int lane  = tid & 31;
    const int wave  = tid >> 5;
    const int t0    = blockIdx.x * 128 + wave * 16;
    const int row   = lane & 15;      // A: M = lane&15 ; D/B: N = lane&15
    const int khalf = lane >> 4;      // A: K-pair select {0,1} vs {2,3}

    const float* arow = visual + ((size_t)b * T_ + t0 + row) * C_ + khalf * 2;
    __builtin_prefetch(arow + 128 * C_, 0, 0);   // hint next t-tile (global_prefetch_b8)

    v8f acc = {0.f, 0.f, 0.f, 0.f, 0.f, 0.f, 0.f, 0.f};
    for (int kc = 0; kc < C_ / 4; ++kc) {
        v2f a, bf;
        a.x  = arow[kc * 4 + 0];                 // K = khalf*2
        a.y  = arow[kc * 4 + 1];                 // K = khalf*2 + 1
        bf.x = wlds[kc * 4 + khalf * 2 + 0][row];
        bf.y = wlds[kc * 4 + khalf * 2 + 1][row];
        acc = __builtin_amdgcn_wmma_f32_16x16x4_f32(
                  false, a, false, bf, (short)0, acc, false, false);
    }

    const int n = lane & 15;                     // output column = head
    if (n < H_) {
        const float bias = qbk[b * H_ + n];
        const float rsD  = 0.088388347648318447f;   // 1/sqrt(128)
        for (int j = 0; j < 8; ++j) {
            int t = t0 + j + khalf * 8;          // D-layout: VGPR j -> M=j / M=j+8
            scores[((size_t)(b * H_ + n)) * T_ + t] = (acc[j] + bias) * rsD;
        }
    }
}

// ------------- Kernel 5: per-(b,h) softmax stats: max, sum(exp) -------------
__global__ void k_stats(const float* __restrict__ scores, float* __restrict__ mx,
                        float* __restrict__ sm) {
    const int bh = blockIdx.x;                   // 64
    const float* s = scores + (size_t)bh * T_;
    __shared__ float red[256];
    int tid = threadIdx.x;
    float m = -1e30f;
    for (int t = tid; t < T_; t += 256) m = fmaxf(m, s[t]);
    red[tid] = m; __syncthreads();
    for (int off = 128; off > 0; off >>= 1) {
        if (tid < off) red[tid] = fmaxf(red[tid], red[tid + off]);
        __syncthreads();
    }
    const float M = red[0];
    __syncthreads();
    float acc = 0.f;
    for (int t = tid; t < T_; t += 256) acc += expf(s[t] - M);
    red[tid] = acc; __syncthreads();
    for (int off = 128; off > 0; off >>= 1) {
        if (tid < off) red[tid] += red[tid + off];
        __syncthreads();
    }
    if (tid == 0) { mx[bh] = M; sm[bh] = red[0]; }
}

// -------- Kernel 6: temp_weights[b,t] = mean_h softmax(scores)[b,h,t] -------
__global__ void k_weights(const float* __restrict__ scores, const float* __restrict__ mx,
                          const float* __restrict__ sm, float* __restrict__ wts) {
    int i = blockIdx.x * blockDim.x + threadIdx.x;   // 0..B_*T_-1
    int b = i / T_, t = i % T_;
    float acc = 0.f;
    for (int h = 0; h < H_; ++h) {
        int bh = b * H_ + h;
        acc += expf(scores[(size_t)bh * T_ + t] - mx[bh]) / sm[bh];
    }
    wts[i] = acc * 0.25f;
}

// ------- Kernel 7: per-batch top-10 (argmax x10), index-sort, gather --------
__global__ void k_topk(const float* __restrict__ wts, const float* __restrict__ audio,
                       const float* __restrict__ pa, const float* __restrict__ pv,
                       float* __restrict__ out) {
    const int b = blockIdx.x;
    __shared__ float w[T_];           // 32 KB (WGP has 320 KB LDS)
    __shared__ float redv[256];
    __shared__ int   redi[256];
    __shared__ int   sel[TOPK];
    const int tid = threadIdx.x;

    for (int t = tid; t < T_; t += 256) w[t] = wts[(size_t)b * T_ + t];
    __syncthreads();

    for (int k = 0; k < TOPK; ++k) {
        float bm = -1e30f; int bi = T_ - 1;
        for (int t = tid; t < T_; t += 256) {
            float v = w[t];
            if (v > bm || (v == bm && t > bi)) { bm = v; bi = t; }  // ties: take larger idx (argsort keeps later dup last)
        }
        redv[tid] = bm; redi[tid] = bi; __syncthreads();
        for (int off = 128; off > 0; off >>= 1) {
            if (tid < off) {
                float v2 = redv[tid + off]; int i2 = redi[tid + off];
                if (v2 > redv[tid] || (v2 == redv[tid] && i2 > redi[tid])) {
                    redv[tid] = v2; redi[tid] = i2;
                }
            }
            __syncthreads();
        }
        if (tid == 0) { sel[k] = redi[0]; w[redi[0]] = -1e30f; }
        __syncthreads();
    }

    if (tid == 0) {   // sort selected indices ascending
        for (int i = 1; i < TOPK; ++i) {
            int v = sel[i], j = i - 1;
            while (j >= 0 && sel[j] > v) { sel[j + 1] = sel[j]; --j; }
            sel[j + 1] = v;
        }
    }
    __syncthreads();

    const size_t SEG = (size_t)B_ * TOPK * C_;   // 81920 floats per output tensor
    for (int i = tid; i < TOPK * C_; i += 256) {
        int k = i / C_, c = i % C_;
        size_t src = ((size_t)b * T_ + sel[k]) * C_ + c;
        size_t dst = ((size_t)b * TOPK + k) * C_ + c;
        out[dst]           = audio[src];
        out[SEG + dst]     = pa[src];
        out[2 * SEG + dst] = pv[src];
    }
}

extern "C" void kernel_launch(void* const* d_in, const int* in_sizes, int n_in,
                              void* d_out, int out_size, void* d_ws, size_t ws_size,
                              hipStream_t stream) {
    const float* audio  = (const float*)d_in[0];
    const float* visual = (const float*)d_in[1];
    const float* pa     = (const float*)d_in[2];
    const float* pv     = (const float*)d_in[3];
    const float* qst    = (const float*)d_in[4];
    const float* Wq     = (const float*)d_in[5];
    const float* bq     = (const float*)d_in[6];
    const float* Wk     = (const float*)d_in[7];
    const float* bk     = (const float*)d_in[8];
    float* out = (float*)d_out;

    // workspace layout (floats)
    float* ws     = (float*)d_ws;
    float* q      = ws;                       // 16*512          = 8192
    float* wmat   = q      + B_ * C_;         // 16*512*4        = 32768
    float* qbk    = wmat   + B_ * C_ * H_;    // 64
    float* scores = qbk    + B_ * H_;         // 16*4*8192       = 524288
    float* mx     = scores + (size_t)B_ * H_ * T_;   // 64
    float* sm     = mx     + B_ * H_;         // 64
    float* wts    = sm     + B_ * H_;         // 16*8192         = 131072
    // total ~ 2.7 MB << ws_size

    k_qproj  <<<(B_ * C_) / 256, 256, 0, stream>>>(qst, Wq, bq, q);
    k_fold   <<<(B_ * C_ * H_) / 256, 256, 0, stream>>>(q, Wk, wmat);
    k_qbk    <<<1, 64, 0, stream>>>(q, bk, qbk);
    k_scores <<<dim3(T_ / 128, B_), 256, 0, stream>>>(visual, wmat, qbk, scores);
    k_stats  <<<B_ * H_, 256, 0, stream>>>(scores, mx, sm);
    k_weights<<<(B_ * T_) / 256, 256, 0, stream>>>(scores, mx, sm, wts);
    k_topk   <<<B_, 256, 0, stream>>>(wts, audio, pa, pv, out);
}